// LocalGConv_74071005986929
// MI455X (gfx1250) — compile-verified
//
#include <hip/hip_runtime.h>

typedef __attribute__((ext_vector_type(2))) float v2f;
typedef __attribute__((ext_vector_type(8))) float v8f;
typedef __attribute__((ext_vector_type(4))) unsigned int v4u;
typedef __attribute__((ext_vector_type(4))) int v4i;
typedef __attribute__((ext_vector_type(8))) int v8i;

#define BN     32768
#define VN     43
#define VP     48
#define DIN_   64
#define DOUT_  64
#define SN     2

// LDS row strides (floats): see bank-conflict reasoning in earlier rounds.
#define ADJ_STR 49
#define W_STR   72     // written by the cooperative W stage: 64 dwords data + 8 dwords pad
#define XB_STR  72
#define TMP_STR 65

#define WAVES   8
#define THREADS (WAVES * 32)
#define GRID    1024

#define XB_TILE (VP * XB_STR)   // floats per x tile (48 rows x 72)

// ---- TDM descriptor helpers (CDNA5 ISA ch.8: D# groups) -------------------
// Group 1 is wave-invariant:
//   word0: data_size=2 (4B) <<16 | pad_enable <<20 | pad_interval=5 (64 dw) <<22
//          | pad_amount=7 (8 dw) <<25
//   tensor_dim0=64 (bits 79:48), tensor_dim1=43 (bits 111:80),
//   tile_dim0=64 (bits 127:112), tile_dim1=48 (bits 143:128),
//   tensor_dim0_stride=64 (bits 207:160). 2D: everything else 0.
__device__ __forceinline__ v8i tdm_group1(void) {
    v8i g;
    g[0] = (int)((2u << 16) | (1u << 20) | (5u << 22) | (7u << 25));
    g[1] = (int)(64u << 16);          // tensor_dim0 low16 @ [31:16]
    g[2] = (int)(43u << 16);          // tensor_dim0 hi16=0, tensor_dim1 low16 @ [31:16]
    g[3] = (int)(64u << 16);          // tensor_dim1 hi16=0, tile_dim0 @ [31:16]
    g[4] = 48;                        // tile_dim1 @ [15:0], tile_dim2=0
    g[5] = 64;                        // tensor_dim0_stride low32
    g[6] = 0;
    g[7] = 0;
    return g;
}

// Group 0: count=1 | lds_addr | 57-bit global byte addr | type=2 @ bits 127:126
__device__ __forceinline__ v4u tdm_group0(unsigned int lds_byte, unsigned long long gaddr) {
    v4u g;
    g[0] = 1u;                                            // count=1, user mode
    g[1] = lds_byte;                                      // LDS dest (bytes)
    g[2] = (unsigned int)gaddr;                           // global addr [31:0]
    g[3] = ((unsigned int)(gaddr >> 32) & 0x01FFFFFFu)    // global addr [56:32]
         | (2u << 30);                                    // type = 2 ("image")
    return g;
}

__device__ __forceinline__ void tdm_load_x(const float* x, int b, unsigned int lds_byte) {
    unsigned long long ga = (unsigned long long)(uintptr_t)(x + (size_t)b * VN * DIN_);
    v4u g0 = tdm_group0(lds_byte, ga);
    v8i g1 = tdm_group1();
    v4i gz4 = {0, 0, 0, 0};
    v8i gz8 = {0, 0, 0, 0, 0, 0, 0, 0};
    __builtin_amdgcn_tensor_load_to_lds(g0, g1, gz4, gz4, gz8, 0);
}
// ---------------------------------------------------------------------------

__global__ __launch_bounds__(THREADS) void
lgconv_wmma_f32(const float* __restrict__ x,
                const float* __restrict__ adj,
                const float* __restrict__ W,
                const float* __restrict__ bias,
                float* __restrict__ out)
{
    __shared__ __align__(16) float adjLds[SN * VP * ADJ_STR];
    __shared__ __align__(16) float wLds[SN * DIN_ * W_STR];
    __shared__ __align__(16) float xbLds[WAVES * 2 * XB_TILE];   // double-buffered
    __shared__ __align__(16) float tmpLds[WAVES * 16 * TMP_STR];

    const int tid = threadIdx.x;

    // Cooperative stage of adj (zero-padded 43 -> 48) and W into LDS.
    for (int i = tid; i < SN * VP * VP; i += THREADS) {
        int s = i / (VP * VP);
        int r = i - s * VP * VP;
        int u = r / VP;
        int v = r - u * VP;
        float val = 0.0f;
        if (u < VN && v < VN) val = adj[(s * VN + u) * VN + v];
        adjLds[(s * VP + u) * ADJ_STR + v] = val;
    }
    for (int i = tid; i < SN * DIN_ * DOUT_; i += THREADS) {
        int s = i / (DIN_ * DOUT_);
        int r = i - s * DIN_ * DOUT_;
        int k = r / DOUT_;
        int o = r - k * DOUT_;
        wLds[(s * DIN_ + k) * W_STR + o] = W[i];
    }
    __syncthreads();

    const int lane = tid & 31;
    const int wave = tid >> 5;
    const int half = lane >> 4;
    const int l15  = lane & 15;

    float bo[4];
    #pragma unroll
    for (int nt = 0; nt < 4; ++nt) bo[nt] = bias[nt * 16 + l15];

    float* xb0 = &xbLds[(wave * 2 + 0) * XB_TILE];
    float* xb1 = &xbLds[(wave * 2 + 1) * XB_TILE];
    float* tmp = &tmpLds[wave * 16 * TMP_STR];

    // LDS byte offsets of the two x tiles (generic-pointer low 32 bits = LDS addr).
    const unsigned int xb_byte0 = (unsigned int)(uintptr_t)(void*)xb0;
    const unsigned int xb_byte1 = (unsigned int)(uintptr_t)(void*)xb1;

    const int gwave  = blockIdx.x * WAVES + wave;
    const int nwaves = gridDim.x * WAVES;

    const v8f vzero = {0.f, 0.f, 0.f, 0.f, 0.f, 0.f, 0.f, 0.f};

    // Prologue: kick off the TDM load of this wave's first batch into buffer 0.
    tdm_load_x(x, gwave, xb_byte0);

    int buf = 0;
    for (int b = gwave; b < BN; b += nwaves) {
        // Issue the next batch's TDM load into the other buffer, then wait for
        // the current tile (TENSORcnt retires in-order, so <=1 means it landed).
        int bn = b + nwaves;
        if (bn < BN) {
            tdm_load_x(x, bn, buf ? xb_byte0 : xb_byte1);
            __builtin_amdgcn_s_wait_tensorcnt(1);
        } else {
            __builtin_amdgcn_s_wait_tensorcnt(0);
        }
        __builtin_amdgcn_wave_barrier();

        const float* xb = buf ? xb1 : xb0;

        for (int mt = 0; mt < 3; ++mt) {
            v8f oacc[4];
            #pragma unroll
            for (int nt = 0; nt < 4; ++nt) oacc[nt] = vzero;

            for (int s = 0; s < SN; ++s) {
                const float* adjS = &adjLds[s * VP * ADJ_STR];
                const float* wS   = &wLds[s * DIN_ * W_STR];

                v8f tacc[4];
                #pragma unroll
                for (int nt = 0; nt < 4; ++nt) tacc[nt] = vzero;

                // Stage A: tmp_tile(16x64) = adj_s[rows mt*16..][0..47] @ x_b(48x64)
                #pragma unroll
                for (int ks = 0; ks < VP / 4; ++ks) {
                    int kb = ks * 4 + 2 * half;
                    v2f a;
                    a[0] = adjS[(mt * 16 + l15) * ADJ_STR + kb];
                    a[1] = adjS[(mt * 16 + l15) * ADJ_STR + kb + 1];
                    #pragma unroll
                    for (int nt = 0; nt < 4; ++nt) {
                        v2f bf;
                        bf[0] = xb[kb * XB_STR + nt * 16 + l15];
                        bf[1] = xb[(kb + 1) * XB_STR + nt * 16 + l15];
                        tacc[nt] = __builtin_amdgcn_wmma_f32_16x16x4_f32(
                            false, a, false, bf, (short)0, tacc[nt], false, false);
                    }
                }

                // C-layout -> A-layout via per-wave LDS tile (DS is in-order per wave).
                __builtin_amdgcn_wave_barrier();
                #pragma unroll
                for (int nt = 0; nt < 4; ++nt)
                    #pragma unroll
                    for (int r = 0; r < 8; ++r)
                        tmp[(r + 8 * half) * TMP_STR + nt * 16 + l15] = tacc[nt][r];
                __builtin_amdgcn_wave_barrier();

                // Stage B: oacc += tmp(16x64) @ W_s(64x64)
                #pragma unroll
                for (int ks = 0; ks < DIN_ / 4; ++ks) {
                    int kb = ks * 4 + 2 * half;
                    v2f a;
                    a[0] = tmp[l15 * TMP_STR + kb];
                    a[1] = tmp[l15 * TMP_STR + kb + 1];
                    #pragma unroll
                    for (int nt = 0; nt < 4; ++nt) {
                        v2f bf;
                        bf[0] = wS[kb * W_STR + nt * 16 + l15];
                        bf[1] = wS[(kb + 1) * W_STR + nt * 16 + l15];
                        oacc[nt] = __builtin_amdgcn_wmma_f32_16x16x4_f32(
                            false, a, false, bf, (short)0, oacc[nt], false, false);
                    }
                }
                __builtin_amdgcn_wave_barrier();   // protect tmp WAR before next support
            }

            // Epilogue: bias + relu, store valid rows only (u < 43).
            #pragma unroll
            for (int nt = 0; nt < 4; ++nt) {
                #pragma unroll
                for (int r = 0; r < 8; ++r) {
                    int u = mt * 16 + r + 8 * half;
                    if (u < VN) {
                        float val = fmaxf(oacc[nt][r] + bo[nt], 0.0f);
                        out[((size_t)b * VN + u) * DOUT_ + nt * 16 + l15] = val;
                    }
                }
            }
        }
        buf ^= 1;
    }
}

extern "C" void kernel_launch(void* const* d_in, const int* in_sizes, int n_in,
                              void* d_out, int out_size, void* d_ws, size_t ws_size,
                              hipStream_t stream) {
    const float* x    = (const float*)d_in[0];   // [32768, 43, 64]
    const float* adj  = (const float*)d_in[1];   // [2, 43, 43]
    const float* W    = (const float*)d_in[2];   // [2, 64, 64]
    const float* bias = (const float*)d_in[3];   // [64]
    float* out        = (float*)d_out;           // [32768, 43, 64]

    lgconv_wmma_f32<<<dim3(GRID), dim3(THREADS), 0, stream>>>(x, adj, W, bias, out);
}